// ChemPropMsgToNode_2319282340444
// MI455X (gfx1250) — compile-verified
//
#include <hip/hip_runtime.h>
#include <hip/hip_bf16.h>

// Problem constants (from the reference)
#define N_NODES 100000
#define N_EDGES 800000
#define D_R     128
#define D_H     96
#define D_K     (D_R + D_H)   // 224
#define D_OUT   96

typedef __attribute__((ext_vector_type(2))) float v2f;
typedef __attribute__((ext_vector_type(8))) float v8f;

// ---------------------------------------------------------------------------
// Stage 0: zero the msg accumulator in workspace
// ---------------------------------------------------------------------------
__global__ void cpm_zero_kernel(float* __restrict__ p, int n) {
    int i = blockIdx.x * blockDim.x + threadIdx.x;
    if (i < n) p[i] = 0.0f;
}

// ---------------------------------------------------------------------------
// Stage 1: scatter-add edge messages onto destination nodes.
// One thread per (edge, channel); hardware f32 atomics (global_atomic_add_f32).
// h reads are fully coalesced; atomics within an edge hit one 96-float row.
// ---------------------------------------------------------------------------
__global__ void cpm_scatter_kernel(const float* __restrict__ h,
                                   const int*   __restrict__ nbrs,
                                   float*       __restrict__ msg) {
    int t = blockIdx.x * blockDim.x + threadIdx.x;   // < 76.8M, fits int32
    if (t >= N_EDGES * D_H) return;
    int e = t / D_H;
    int c = t - e * D_H;
    int dst = nbrs[2 * e];                 // nbrs[:,0] = destination node
    unsafeAtomicAdd(&msg[dst * D_H + c], h[t]);
}

// ---------------------------------------------------------------------------
// Stage 2: fused concat + GEMM + ReLU using V_WMMA_F32_16X16X4_F32.
//   out[m,n] = relu( sum_k cat[m,k] * W[n,k] ),  cat = [r | msg], K=224.
// Block = 192 threads = 6 waves; block owns one 16-row M-tile, each wave owns
// one 16-col N-tile. A-tile staged in LDS (stride 226 dwords: 8B-aligned b64,
// bank-conflict-free across the 16 rows). W stays in global (86KB, L2-resident).
//
// f32 WMMA operand layout (ISA 7.12.2):
//   A 16x4 : lanes 0-15 -> M=lane,  V0=K0, V1=K1 ; lanes 16-31 -> M=lane-16, V0=K2, V1=K3
//   B 4x16 : lanes 0-15 -> N=lane,  V0=K0, V1=K1 ; lanes 16-31 -> N=lane-16, V0=K2, V1=K3
//   C/D    : lanes 0-15 -> M=v (v=VGPR 0..7), lanes 16-31 -> M=v+8 ; N=lane&15
// ---------------------------------------------------------------------------
#define A_STRIDE 226   // 224 padded: even (b64 align) and conflict-free

__global__ __launch_bounds__(192)
void cpm_gemm_kernel(const float* __restrict__ r,
                     const float* __restrict__ msg,
                     const float* __restrict__ W,     // [D_OUT][D_K] row-major
                     float*       __restrict__ out) { // [N_NODES][D_OUT]
    __shared__ __attribute__((aligned(16))) float At[16 * A_STRIDE];

    const int tid    = threadIdx.x;          // 0..191
    const int m_base = blockIdx.x * 16;

    // Stage A-tile: 16 rows x 224 cols of [r | msg] -> LDS (ds_store)
    for (int i = tid; i < 16 * D_K; i += 192) {
        int row = i / D_K;
        int k   = i - row * D_K;
        int m   = m_base + row;
        float v = (k < D_R) ? r[m * D_R + k]
                            : msg[m * D_H + (k - D_R)];
        At[row * A_STRIDE + k] = v;
    }
    __syncthreads();

    const int wave   = tid >> 5;             // 0..5 -> N-tile
    const int lane   = tid & 31;
    const int half   = lane >> 4;            // 0: K+0/1, 1: K+2/3
    const int l15    = lane & 15;
    const int n_base = wave * 16;
    const int koff   = half * 2;

    const float* arow = &At[l15 * A_STRIDE + koff];
    const float* brow = &W[(n_base + l15) * D_K + koff];   // B[k][n] = W[n][k]

    v8f acc = {};
    #pragma unroll 8
    for (int k0 = 0; k0 < D_K; k0 += 4) {
        v2f a = *(const v2f*)(arow + k0);        // ds_load_b64
        v2f b = *(const v2f*)(brow + k0);        // global_load_b64 (L2 hit)
        // 8 args: (neg_a, A, neg_b, B, c_mod, C, reuse_a, reuse_b)
        acc = __builtin_amdgcn_wmma_f32_16x16x4_f32(
            false, a, false, b, (short)0, acc, false, false);
    }

    // ReLU + store per the 16x16 f32 C/D layout
    const int n = n_base + l15;
    #pragma unroll
    for (int v = 0; v < 8; ++v) {
        int m   = m_base + v + half * 8;
        float x = acc[v];
        out[m * D_OUT + n] = x > 0.0f ? x : 0.0f;
    }
}

// ---------------------------------------------------------------------------
// Launch
// ---------------------------------------------------------------------------
extern "C" void kernel_launch(void* const* d_in, const int* in_sizes, int n_in,
                              void* d_out, int out_size, void* d_ws, size_t ws_size,
                              hipStream_t stream) {
    const float* r    = (const float*)d_in[0];   // [N_NODES, D_R]
    const float* h    = (const float*)d_in[1];   // [N_EDGES, D_H]
    const int*   nbrs = (const int*)  d_in[2];   // [N_EDGES, 2]
    const float* W    = (const float*)d_in[3];   // [D_OUT, D_K]
    float*       out  = (float*)d_out;           // [N_NODES, D_OUT]
    float*       msgb = (float*)d_ws;            // [N_NODES, D_H] scratch

    const int nmsg = N_NODES * D_H;              // 9.6M floats
    cpm_zero_kernel<<<(nmsg + 255) / 256, 256, 0, stream>>>(msgb, nmsg);

    const int nscat = N_EDGES * D_H;             // 76.8M
    cpm_scatter_kernel<<<(nscat + 255) / 256, 256, 0, stream>>>(h, nbrs, msgb);

    cpm_gemm_kernel<<<N_NODES / 16, 192, 0, stream>>>(r, msgb, W, out);
}